// BatchedBjorckOrthogonalization_45122926411979
// MI455X (gfx1250) — compile-verified
//
#include <hip/hip_runtime.h>

// Batched Björck orthogonalization, 8192 x (64x64) fp32, 13 iterations of
//   S = W^T W ;  W = 1.5*W - 0.5*(W*S)
// One workgroup per matrix, matrix resident in LDS across all iterations.
// fp32 WMMA (v_wmma_f32_16x16x4_f32). Both W row-major and W-transposed are
// kept in LDS so every WMMA fragment is a single contiguous ds_load_b64, and
// S's symmetry lets GEMM2 read row-major S as contiguous K-pairs too.

typedef __attribute__((ext_vector_type(2))) float v2f;
typedef __attribute__((ext_vector_type(4))) float v4f;
typedef __attribute__((ext_vector_type(8))) float v8f;

#define NDIM    64
#define LS      68      // padded LDS row stride in floats (rows 16B-aligned)
#define ITERS   13      // BACKPROP_ITERS + NON_BACKPROP_ITERS
#define THREADS 256     // 8 wave32

__global__ __launch_bounds__(THREADS)
void bjorck_wmma_kernel(const float* __restrict__ w_in, float* __restrict__ w_out) {
  __shared__ float Wl [NDIM * LS];  // W, row-major
  __shared__ float Wtl[NDIM * LS];  // W, transposed
  __shared__ float Sl [NDIM * LS];  // S = W^T W (symmetric; row n holds S[n][*])

  const unsigned tid  = threadIdx.x;
  const unsigned lane = tid & 31u;
  const unsigned wave = tid >> 5;          // 0..7
  const unsigned half = lane >> 4;         // K-pair / M-half select
  const unsigned colm = lane & 15u;        // M (A) or N (B/C) within tile
  const unsigned ti   = wave >> 1;         // tile row 0..3
  const unsigned tj0  = (wave & 1u) << 1;  // tile cols {tj0, tj0+1}

  // ---------------- global -> LDS (Wl direct, Wtl transposed) ----------------
  const float4* gin = (const float4*)(w_in + (size_t)blockIdx.x * (NDIM * NDIM));
  #pragma unroll
  for (int i = 0; i < 4; ++i) {
    unsigned c = tid + THREADS * i;        // 1024 float4 chunks: 64 rows x 16
    unsigned row = c >> 4, q = c & 15u;
    float4 v = gin[c];
    *(float4*)&Wl[row * LS + q * 4u] = v;
    Wtl[(q * 4u + 0u) * LS + row] = v.x;
    Wtl[(q * 4u + 1u) * LS + row] = v.y;
    Wtl[(q * 4u + 2u) * LS + row] = v.z;
    Wtl[(q * 4u + 3u) * LS + row] = v.w;
  }
  __syncthreads();

  // Loop-invariant per-lane fragment base pointers (immediate-offset friendly).
  const float* aT  = &Wtl[(16u * ti         + colm) * LS + 2u * half]; // GEMM1 A
  const float* b0T = &Wtl[(16u * tj0        + colm) * LS + 2u * half]; // GEMM1 B0
  const float* b1T = &Wtl[(16u * (tj0 + 1u) + colm) * LS + 2u * half]; // GEMM1 B1
  const float* aR  = &Wl [(16u * ti         + colm) * LS + 2u * half]; // GEMM2 A
  const float* s0R = &Sl [(16u * tj0        + colm) * LS + 2u * half]; // GEMM2 B0
  const float* s1R = &Sl [(16u * (tj0 + 1u) + colm) * LS + 2u * half]; // GEMM2 B1
  // Transposed-coordinate tile bases (contiguous along M for this lane).
  float* st0 = &Sl [(16u * tj0        + colm) * LS + 16u * ti + 8u * half];
  float* st1 = &Sl [(16u * (tj0 + 1u) + colm) * LS + 16u * ti + 8u * half];
  float* wt0 = &Wtl[(16u * tj0        + colm) * LS + 16u * ti + 8u * half];
  float* wt1 = &Wtl[(16u * (tj0 + 1u) + colm) * LS + 16u * ti + 8u * half];

  #pragma unroll 1
  for (int it = 0; it < ITERS; ++it) {
    // ---------------- S = W^T * W (all operands from Wt, b64 loads) ----------
    v8f s0 = {}, s1 = {};
    #pragma unroll
    for (unsigned kb = 0; kb < 16; ++kb) {
      v2f a  = *(const v2f*)(aT  + 4u * kb);
      v2f b0 = *(const v2f*)(b0T + 4u * kb);
      v2f b1 = *(const v2f*)(b1T + 4u * kb);
      s0 = __builtin_amdgcn_wmma_f32_16x16x4_f32(false, a, false, b0, (short)0, s0, false, false);
      s1 = __builtin_amdgcn_wmma_f32_16x16x4_f32(false, a, false, b1, (short)0, s1, false, false);
    }
    // S is symmetric: store tile (ti,tj) at transposed coords -> b128 stores.
    {
      v4f lo0 = {s0[0], s0[1], s0[2], s0[3]}, hi0 = {s0[4], s0[5], s0[6], s0[7]};
      v4f lo1 = {s1[0], s1[1], s1[2], s1[3]}, hi1 = {s1[4], s1[5], s1[6], s1[7]};
      *(v4f*)(st0)      = lo0;  *(v4f*)(st0 + 4) = hi0;
      *(v4f*)(st1)      = lo1;  *(v4f*)(st1 + 4) = hi1;
    }
    __syncthreads();  // S fully built

    // ------------- P = W * S ; fused Wnew = 1.5W - 0.5P ---------------------
    v8f p0 = {}, p1 = {};
    #pragma unroll
    for (unsigned kb = 0; kb < 16; ++kb) {
      v2f a  = *(const v2f*)(aR  + 4u * kb);
      v2f b0 = *(const v2f*)(s0R + 4u * kb);   // S[n][k] == S[k][n]
      v2f b1 = *(const v2f*)(s1R + 4u * kb);
      p0 = __builtin_amdgcn_wmma_f32_16x16x4_f32(false, a, false, b0, (short)0, p0, false, false);
      p1 = __builtin_amdgcn_wmma_f32_16x16x4_f32(false, a, false, b1, (short)0, p1, false, false);
    }
    // Old W tile via Wt (contiguous b128 reads), fused AXPY.
    v4f w0lo = *(const v4f*)(wt0), w0hi = *(const v4f*)(wt0 + 4);
    v4f w1lo = *(const v4f*)(wt1), w1hi = *(const v4f*)(wt1 + 4);
    v4f n0lo, n0hi, n1lo, n1hi;
    #pragma unroll
    for (int r = 0; r < 4; ++r) {
      n0lo[r] = 1.5f * w0lo[r] - 0.5f * p0[r];
      n0hi[r] = 1.5f * w0hi[r] - 0.5f * p0[r + 4];
      n1lo[r] = 1.5f * w1lo[r] - 0.5f * p1[r];
      n1hi[r] = 1.5f * w1hi[r] - 0.5f * p1[r + 4];
    }
    __syncthreads();  // all reads of Wl/Wtl/Sl complete before overwrite

    // Write Wnew: Wt gets contiguous b128 stores, Wl gets per-row b32 stores.
    *(v4f*)(wt0)      = n0lo;  *(v4f*)(wt0 + 4) = n0hi;
    *(v4f*)(wt1)      = n1lo;  *(v4f*)(wt1 + 4) = n1hi;
    #pragma unroll
    for (unsigned r = 0; r < 8; ++r) {
      unsigned row = 16u * ti + 8u * half + r;
      float v0 = (r < 4u) ? n0lo[r] : n0hi[r - 4u];
      float v1 = (r < 4u) ? n1lo[r] : n1hi[r - 4u];
      Wl[row * LS + 16u * tj0 + colm]        = v0;
      Wl[row * LS + 16u * (tj0 + 1u) + colm] = v1;
    }
    __syncthreads();  // new W visible to everyone
  }

  // ---------------- LDS -> global (float4 vectorized) ----------------
  float4* gout = (float4*)(w_out + (size_t)blockIdx.x * (NDIM * NDIM));
  #pragma unroll
  for (int i = 0; i < 4; ++i) {
    unsigned c = tid + THREADS * i;
    unsigned row = c >> 4, q = c & 15u;
    gout[c] = *(const float4*)&Wl[row * LS + q * 4u];
  }
}

extern "C" void kernel_launch(void* const* d_in, const int* in_sizes, int n_in,
                              void* d_out, int out_size, void* d_ws, size_t ws_size,
                              hipStream_t stream) {
  const float* w = (const float*)d_in[0];
  float* out = (float*)d_out;
  const int batch = in_sizes[0] / (NDIM * NDIM);  // 8192
  bjorck_wmma_kernel<<<batch, THREADS, 0, stream>>>(w, out);
}